// GNNModel_76261439308343
// MI455X (gfx1250) — compile-verified
//
#include <hip/hip_runtime.h>

typedef float v2f __attribute__((ext_vector_type(2)));
typedef float v4f __attribute__((ext_vector_type(4)));
typedef float v8f __attribute__((ext_vector_type(8)));

#define WMMA_F32_4(a, b, c) \
  __builtin_amdgcn_wmma_f32_16x16x4_f32(false, (a), false, (b), (short)0, (c), false, false)

// ---------------------------------------------------------------------------
// K1: xw1 = x @ W1            [N,128] @ [128,2] -> [N,2]
// ---------------------------------------------------------------------------
__global__ void gnn_k1_xw1(const float* __restrict__ x, const float* __restrict__ W1,
                           float* __restrict__ xw1, int n, int f) {
  int row = blockIdx.x * blockDim.x + threadIdx.x;
  if (row >= n) return;
  const float* xr = x + (size_t)row * f;
  float a0 = 0.f, a1 = 0.f;
  for (int j = 0; j < f; j += 4) {
    v4f xv = *(const v4f*)(xr + j);
    a0 += xv.x * W1[2*j    ] + xv.y * W1[2*j + 2] + xv.z * W1[2*j + 4] + xv.w * W1[2*j + 6];
    a1 += xv.x * W1[2*j + 1] + xv.y * W1[2*j + 3] + xv.z * W1[2*j + 5] + xv.w * W1[2*j + 7];
  }
  xw1[2*row]     = a0;
  xw1[2*row + 1] = a1;
}

// ---------------------------------------------------------------------------
// K2: h1 = relu(A @ xw1 + b1) ; g = h1 @ W2   (fused). g stored PRE-PACKED in
// WMMA B-fragment order: g2[(r>>1)*32 + col*2 + (r&1)] = g[r][col]
// One 256-thread block per row streams the 80 KB A row (HBM pass #1, 1 FLOP/B).
// ---------------------------------------------------------------------------
__global__ void gnn_k2_layer1(const float* __restrict__ A, const float* __restrict__ xw1,
                              const float* __restrict__ b1, const float* __restrict__ W2,
                              float* __restrict__ g2, int n) {
  const int row = blockIdx.x;
  const int tid = threadIdx.x;
  const float* ar = A + (size_t)row * n;
  float a0 = 0.f, a1 = 0.f;
  const int nq = n >> 2;
  for (int q = tid; q < nq; q += 256) {
    v4f av = *(const v4f*)(ar + 4 * q);          // 16B of A
    v4f w0 = *(const v4f*)(xw1 + 8 * q);         // xw1 pairs (L2-hot)
    v4f w1 = *(const v4f*)(xw1 + 8 * q + 4);
    a0 += av.x * w0.x + av.y * w0.z + av.z * w1.x + av.w * w1.z;
    a1 += av.x * w0.y + av.y * w0.w + av.z * w1.y + av.w * w1.w;
  }
  for (int off = 16; off > 0; off >>= 1) {       // wave32 reduction
    a0 += __shfl_xor(a0, off, 32);
    a1 += __shfl_xor(a1, off, 32);
  }
  __shared__ float wsum[8][2];
  __shared__ float h1s[2];
  const int wave = tid >> 5, lane = tid & 31;
  if (lane == 0) { wsum[wave][0] = a0; wsum[wave][1] = a1; }
  __syncthreads();
  if (tid == 0) {
    float s0 = 0.f, s1 = 0.f;
    for (int w = 0; w < 8; ++w) { s0 += wsum[w][0]; s1 += wsum[w][1]; }
    s0 += b1[0]; s1 += b1[1];
    h1s[0] = s0 > 0.f ? s0 : 0.f;
    h1s[1] = s1 > 0.f ? s1 : 0.f;
  }
  __syncthreads();
  if (tid < 16) {
    float gv = h1s[0] * W2[tid] + h1s[1] * W2[16 + tid];
    g2[(size_t)(row >> 1) * 32 + tid * 2 + (row & 1)] = gv;   // B-fragment packing
  }
}

// ---------------------------------------------------------------------------
// K3: h2 = relu(A @ g + b2) ; out = h2 @ Wd + bd   (fused)
// HBM pass #2 at 8 FLOP/B -> V_WMMA_F32_16X16X4_F32 (2048 FLOP per 256B of A).
// Block = 16 output rows, 8 waves interleave over K with stride 32 so the
// block consumes full 128B cachelines per row per step. UNIFORM scalar trip
// count (n>>5) so no exec masking around WMMA; manual 4x unroll with two
// independent accumulators so loads of a group overlap the WMMA chain.
// ---------------------------------------------------------------------------
__global__ void gnn_k3_layer2(const float* __restrict__ A, const float* __restrict__ g2,
                              const float* __restrict__ b2, const float* __restrict__ Wd,
                              const float* __restrict__ bd, float* __restrict__ out, int n) {
  const int tid  = threadIdx.x;
  const int wave = tid >> 5;
  const int lane = tid & 31;
  const int lo   = lane & 15;            // M for A-frag, Ncol for B-frag
  const int koff = (lane >> 4) << 1;     // lanes 0-15 -> K{0,1}, lanes 16-31 -> K{2,3}
  const int row0 = blockIdx.x << 4;

  // per-lane streaming pointers; constant strides fold into load offsets
  const float* ap = A  + (size_t)(row0 + lo) * n + (size_t)(wave * 4 + koff);
  const float* bp = g2 + (size_t)(wave * 4 + koff) * 16 + lo * 2;   // packed B

  v8f acc0 = {}, acc1 = {};
  const int niters = n >> 5;             // uniform: each wave does n/32 K-steps
  int it = 0;
  for (; it + 4 <= niters; it += 4) {
    v2f a0 = *(const v2f*)(ap);        v2f b0v = *(const v2f*)(bp);
    v2f a1 = *(const v2f*)(ap + 32);   v2f b1v = *(const v2f*)(bp + 512);
    v2f a2 = *(const v2f*)(ap + 64);   v2f b2v = *(const v2f*)(bp + 1024);
    v2f a3 = *(const v2f*)(ap + 96);   v2f b3v = *(const v2f*)(bp + 1536);
    acc0 = WMMA_F32_4(a0, b0v, acc0);
    acc1 = WMMA_F32_4(a1, b1v, acc1);
    acc0 = WMMA_F32_4(a2, b2v, acc0);
    acc1 = WMMA_F32_4(a3, b3v, acc1);
    ap += 128;            // 4 K-steps * 32 columns
    bp += 2048;           // 4 K-steps * 32 rows * 16 cols (packed)
  }
  for (; it < niters; ++it) {
    v2f a = *(const v2f*)(ap);
    v2f b = *(const v2f*)(bp);
    acc0 = WMMA_F32_4(a, b, acc0);
    ap += 32;
    bp += 512;
  }

  // reduce 8 per-wave partial C tiles (merging acc0+acc1 on the way in)
  __shared__ float red[8 * 256];
#pragma unroll
  for (int v = 0; v < 8; ++v) red[wave * 256 + v * 32 + lane] = acc0[v] + acc1[v];
  __syncthreads();

  float s = 0.f;
#pragma unroll
  for (int w = 0; w < 8; ++w) s += red[w * 256 + tid];

  // C layout: VGPR v, lane l -> (m = v + 8*(l>=16), ncol = l&15)
  const int cv = tid >> 5;
  const int cl = tid & 31;
  const int m  = cv + ((cl & 16) ? 8 : 0);
  const int nc = cl & 15;
  float h2 = s + b2[nc];
  h2 = h2 > 0.f ? h2 : 0.f;
  __syncthreads();
  red[m * 16 + nc] = h2 * Wd[nc];
  __syncthreads();
  if (tid < 16) {
    float o = bd[0];
#pragma unroll
    for (int q = 0; q < 16; ++q) o += red[tid * 16 + q];
    out[row0 + tid] = o;
  }
}

// ---------------------------------------------------------------------------
extern "C" void kernel_launch(void* const* d_in, const int* in_sizes, int n_in,
                              void* d_out, int out_size, void* d_ws, size_t ws_size,
                              hipStream_t stream) {
  const float* x  = (const float*)d_in[0];
  const float* A  = (const float*)d_in[1];
  const float* W1 = (const float*)d_in[2];
  const float* b1 = (const float*)d_in[3];
  const float* W2 = (const float*)d_in[4];
  const float* b2 = (const float*)d_in[5];
  const float* Wd = (const float*)d_in[6];
  const float* bd = (const float*)d_in[7];
  float* out = (float*)d_out;

  const int n = out_size;              // 20000 (divisible by 32)
  const int f = in_sizes[0] / n;       // 128

  float* xw1 = (float*)d_ws;                              // n*2 floats
  size_t g2_off = ((size_t)(2 * n) + 63) & ~(size_t)63;   // 64-float align
  float* g2 = (float*)d_ws + g2_off;                      // n*16 floats, B-packed

  gnn_k1_xw1   <<<(n + 255) / 256, 256, 0, stream>>>(x, W1, xw1, n, f);
  gnn_k2_layer1<<<n,               256, 0, stream>>>(A, xw1, b1, W2, g2, n);
  gnn_k3_layer2<<<n / 16,          256, 0, stream>>>(A, g2, b2, Wd, bd, out, n);
}